// TemporalSampling_62534723829926
// MI455X (gfx1250) — compile-verified
//
#include <hip/hip_runtime.h>
#include <math.h>

// Problem constants (fixed by reference setup_inputs)
#define C_DIM   128
#define HH      96
#define WW      96
#define T_DIM   (HH*WW)      // 9216
#define DSTATE  8
#define NCHUNK  72
#define CHUNK   128          // NCHUNK*CHUNK == T_DIM

typedef __attribute__((ext_vector_type(2))) float v2f;
typedef __attribute__((ext_vector_type(8))) float v8f;

__device__ __forceinline__ float sigmoidf_(float x){ return 1.f/(1.f+__expf(-x)); }
__device__ __forceinline__ float siluf_(float x){ return x*sigmoidf_(x); }
__device__ __forceinline__ float softplusf_(float x){ return fmaxf(x,0.f)+log1pf(__expf(-fabsf(x))); }

// ---------------------------------------------------------------------------
// K0: precompute (single block):
//   active[n]      : 0 for group-start (ego) elements, 1 otherwise
//   Af/Ab          : -exp(A_log)
//   Wip[128][256]  : diag(rms_w) @ in_proj      (rmsnorm weight folded in)
//   Wc [128][144]  : [x_proj[:, :8]@dt_proj | x_proj[:,8:24]]
//   w2 [128]       : out_proj_w @ mlp_w
// ---------------------------------------------------------------------------
__global__ void k0_precomp(const int* __restrict__ rec, int ngroups, int N,
                           const float* __restrict__ in_proj,
                           const float* __restrict__ rms_w,
                           const float* __restrict__ x_proj_w,
                           const float* __restrict__ dt_proj_w,
                           const float* __restrict__ out_proj_w,
                           const float* __restrict__ mlp_w,
                           const float* __restrict__ A_log_f,
                           const float* __restrict__ A_log_b,
                           int* __restrict__ active,
                           float* __restrict__ Wip, float* __restrict__ Wc,
                           float* __restrict__ w2,
                           float* __restrict__ Af, float* __restrict__ Ab)
{
    int tid = threadIdx.x;
    if (tid == 0) {
        for (int n = 0; n < N; ++n) active[n] = 1;
        int cum = 0;
        for (int g = 0; g < ngroups; ++g) {
            if (cum < N) active[cum] = 0;   // first element of each group -> ego
            cum += rec[g];
        }
    }
    for (int i = tid; i < C_DIM*DSTATE; i += blockDim.x) {
        Af[i] = -__expf(A_log_f[i]);
        Ab[i] = -__expf(A_log_b[i]);
    }
    for (int i = tid; i < C_DIM*2*C_DIM; i += blockDim.x) {
        int c = i >> 8;                      // 256 cols
        Wip[i] = rms_w[c]*in_proj[i];
    }
    for (int i = tid; i < C_DIM*144; i += blockDim.x) {
        int c = i/144, j = i - c*144;
        float v;
        if (j < 128) { v = 0.f; for (int r = 0; r < 8; ++r) v += x_proj_w[c*24+r]*dt_proj_w[r*128+j]; }
        else v = x_proj_w[c*24 + 8 + (j-128)];
        Wc[i] = v;
    }
    for (int c = tid; c < C_DIM; c += blockDim.x) {
        float v = 0.f;
        for (int j = 0; j < C_DIM; ++j) v += out_proj_w[c*C_DIM+j]*mlp_w[j];
        w2[c] = v;
    }
}

// ---------------------------------------------------------------------------
// K1: rmsnorm row scales  s[t] = rsqrt(mean_c feats[c][t]^2 + eps)
// ---------------------------------------------------------------------------
__global__ void k1_rms(const float* __restrict__ feats, const int* __restrict__ active,
                       float* __restrict__ s)
{
    int e = blockIdx.z; if (!active[e]) return;
    int t = blockIdx.x*blockDim.x + threadIdx.x; if (t >= T_DIM) return;
    const float* f = feats + (size_t)e*C_DIM*T_DIM + t;
    float acc = 0.f;
    for (int c = 0; c < C_DIM; ++c) { float v = f[(size_t)c*T_DIM]; acc += v*v; }
    s[(size_t)e*T_DIM + t] = rsqrtf(acc*(1.f/(float)C_DIM) + 1e-5f);
}

// ---------------------------------------------------------------------------
// K2: GEMM1  xz[t][j] = s[t] * sum_c feats[c][t] * Wip[c][j]   (j in [0,256))
//   f32 WMMA 16x16x4. feats is GEMM-A column-major (lane = M) -> coalesced.
//   Each wave owns one 16-col j-tile, caches all 32 B fragments in registers,
//   streams 16 t-tiles. grid(4, 36, N), block 128 (4 waves).
// ---------------------------------------------------------------------------
__global__ void k2_gemm1(const float* __restrict__ feats, const float* __restrict__ Wip,
                         const float* __restrict__ s, const int* __restrict__ active,
                         float* __restrict__ xi0, float* __restrict__ z)
{
    int e = blockIdx.z; if (!active[e]) return;
    int wave = threadIdx.x >> 5, lane = threadIdx.x & 31;
    int half = lane >> 4, l = lane & 15;
    int j0 = (blockIdx.x*4 + wave)*16;         // 0..240

    v2f bfrag[32];
    #pragma unroll
    for (int k = 0; k < 32; ++k) {
        const float* bp = Wip + (size_t)(k*4 + 2*half)*256 + j0 + l;
        bfrag[k].x = bp[0];
        bfrag[k].y = bp[256];
    }

    const float* fb = feats + (size_t)e*C_DIM*T_DIM;
    const float* se = s + (size_t)e*T_DIM;
    float* xie = xi0 + (size_t)e*T_DIM*C_DIM;
    float* ze  = z   + (size_t)e*T_DIM*C_DIM;
    bool isXi = (j0 < 128);

    for (int it = 0; it < 16; ++it) {
        int t0 = (blockIdx.y*16 + it)*16;
        int trow = t0 + l;
        v8f acc = {0.f,0.f,0.f,0.f,0.f,0.f,0.f,0.f};
        #pragma unroll
        for (int k = 0; k < 32; ++k) {
            int ka = k*4 + 2*half;
            v2f a;
            a.x = fb[(size_t)ka    *T_DIM + trow];
            a.y = fb[(size_t)(ka+1)*T_DIM + trow];
            acc = __builtin_amdgcn_wmma_f32_16x16x4_f32(false, a, false, bfrag[k],
                                                        (short)0, acc, false, false);
        }
        #pragma unroll
        for (int r = 0; r < 8; ++r) {
            int t = t0 + r + 8*half;
            float v = acc[r]*se[t];
            if (isXi) xie[(size_t)t*C_DIM + (j0 + l)]        = v;
            else      ze [(size_t)t*C_DIM + (j0 + l - 128)]  = v;
        }
    }
}

// ---------------------------------------------------------------------------
// K3: depthwise causal conv (k=4, left pad 3) + SiLU.  thread = channel.
// ---------------------------------------------------------------------------
__global__ void k3_conv(const float* __restrict__ xi0, const float* __restrict__ conv_w,
                        const float* __restrict__ conv_b, const int* __restrict__ active,
                        float* __restrict__ xi)
{
    int e = blockIdx.z; if (!active[e]) return;
    int c = threadIdx.x;
    const float* src = xi0 + (size_t)e*T_DIM*C_DIM + c;
    float*       dst = xi  + (size_t)e*T_DIM*C_DIM + c;
    int t0 = blockIdx.x * CHUNK;
    float w0 = conv_w[c*4+0], w1 = conv_w[c*4+1], w2 = conv_w[c*4+2], w3 = conv_w[c*4+3];
    float b = conv_b[c];
    float x0 = (t0-3 >= 0) ? src[(size_t)(t0-3)*C_DIM] : 0.f;
    float x1 = (t0-2 >= 0) ? src[(size_t)(t0-2)*C_DIM] : 0.f;
    float x2 = (t0-1 >= 0) ? src[(size_t)(t0-1)*C_DIM] : 0.f;
    for (int i = 0; i < CHUNK; ++i) {
        int t = t0 + i;
        float x3 = src[(size_t)t*C_DIM];
        float y = b + w0*x0 + w1*x1 + w2*x2 + w3*x3;
        dst[(size_t)t*C_DIM] = siluf_(y);
        x0 = x1; x1 = x2; x2 = x3;
    }
}

// ---------------------------------------------------------------------------
// K4: GEMM2  xi(Tx128) @ Wc(128x144) -> delta (softplus(.+dt_b), cols 0..127)
//            and B/C (cols 128..143).  grid(9, 36, N), block 32 (1 wave).
// ---------------------------------------------------------------------------
__global__ void k4_gemm2(const float* __restrict__ xi, const float* __restrict__ Wc,
                         const float* __restrict__ dt_b, const int* __restrict__ active,
                         float* __restrict__ delta, float* __restrict__ bc)
{
    int e = blockIdx.z; if (!active[e]) return;
    int lane = threadIdx.x & 31;
    int half = lane >> 4, l = lane & 15;
    int j0 = blockIdx.x*16;                     // 0..128

    v2f bfrag[32];
    #pragma unroll
    for (int k = 0; k < 32; ++k) {
        const float* bp = Wc + (size_t)(k*4 + 2*half)*144 + j0 + l;
        bfrag[k].x = bp[0];
        bfrag[k].y = bp[144];
    }

    const float* xie = xi + (size_t)e*T_DIM*C_DIM;
    float* de  = delta + (size_t)e*T_DIM*C_DIM;
    float* bce = bc    + (size_t)e*T_DIM*16;
    bool isDelta = (j0 < 128);
    float bias = isDelta ? dt_b[j0 + l] : 0.f;

    for (int it = 0; it < 16; ++it) {
        int t0 = (blockIdx.y*16 + it)*16;
        int trow = t0 + l;
        v8f acc = {0.f,0.f,0.f,0.f,0.f,0.f,0.f,0.f};
        #pragma unroll
        for (int k = 0; k < 32; ++k) {
            v2f a = *(const v2f*)(xie + (size_t)trow*C_DIM + k*4 + 2*half);
            acc = __builtin_amdgcn_wmma_f32_16x16x4_f32(false, a, false, bfrag[k],
                                                        (short)0, acc, false, false);
        }
        #pragma unroll
        for (int r = 0; r < 8; ++r) {
            int t = t0 + r + 8*half;
            float v = acc[r];
            if (isDelta) de [(size_t)t*C_DIM + (j0 + l)]      = softplusf_(v + bias);
            else         bce[(size_t)t*16   + (j0 + l - 128)] = v;
        }
    }
}

// ---------------------------------------------------------------------------
// K5a: chunked scan pass A: per (chunk, dir, channel) compute per-state
//      (prod dA, h_partial from h0=0) over 128 steps. grid(72, 2, N), block 128.
// ---------------------------------------------------------------------------
__global__ void k5a_part(const float* __restrict__ delta, const float* __restrict__ xi,
                         const float* __restrict__ bc,
                         const float* __restrict__ Af, const float* __restrict__ Ab,
                         const int* __restrict__ active, float* __restrict__ part)
{
    int e = blockIdx.z; if (!active[e]) return;
    int dir = blockIdx.y, ch = blockIdx.x, d = threadIdx.x;
    const float* A = (dir == 0 ? Af : Ab) + d*DSTATE;
    float Ar[DSTATE];
    #pragma unroll
    for (int n = 0; n < DSTATE; ++n) Ar[n] = A[n];
    const float* de  = delta + (size_t)e*T_DIM*C_DIM;
    const float* xe  = xi    + (size_t)e*T_DIM*C_DIM;
    const float* bce = bc    + (size_t)e*T_DIM*16;
    float h[DSTATE]  = {0.f,0.f,0.f,0.f,0.f,0.f,0.f,0.f};
    float ap[DSTATE] = {1.f,1.f,1.f,1.f,1.f,1.f,1.f,1.f};
    for (int i = 0; i < CHUNK; ++i) {
        int p = ch*CHUNK + i;
        int t = dir ? (T_DIM-1-p) : p;
        float dl = de[(size_t)t*C_DIM + d];
        float xv = xe[(size_t)t*C_DIM + d];
        float dx = dl*xv;
        const float* bcr = bce + (size_t)t*16;
        #pragma unroll
        for (int n = 0; n < DSTATE; ++n) {
            float da = __expf(dl*Ar[n]);
            h[n]  = da*h[n] + dx*bcr[n];
            ap[n] *= da;
        }
    }
    float* pp = part + ((((size_t)e*2 + dir)*NCHUNK + ch)*C_DIM + d)*16;
    #pragma unroll
    for (int n = 0; n < DSTATE; ++n) { pp[n] = ap[n]; pp[8+n] = h[n]; }
}

// K5b: thread the 72 chunk carries. grid(2, N), block 128.
__global__ void k5b_carry(const int* __restrict__ active, const float* __restrict__ part,
                          float* __restrict__ h0)
{
    int e = blockIdx.y; if (!active[e]) return;
    int dir = blockIdx.x, d = threadIdx.x;
    float carry[DSTATE] = {0.f,0.f,0.f,0.f,0.f,0.f,0.f,0.f};
    for (int ch = 0; ch < NCHUNK; ++ch) {
        const float* pp = part + ((((size_t)e*2 + dir)*NCHUNK + ch)*C_DIM + d)*16;
        float*       hh = h0   + ((((size_t)e*2 + dir)*NCHUNK + ch)*C_DIM + d)*8;
        #pragma unroll
        for (int n = 0; n < DSTATE; ++n) {
            hh[n] = carry[n];
            carry[n] = pp[n]*carry[n] + pp[8+n];
        }
    }
}

// K5c: rescan with correct h0, emit y = sum_n h*C + x*D.  grid(72, 2, N), block 128.
__global__ void k5c_scan(const float* __restrict__ delta, const float* __restrict__ xi,
                         const float* __restrict__ bc,
                         const float* __restrict__ Af, const float* __restrict__ Ab,
                         const float* __restrict__ Df, const float* __restrict__ Db,
                         const float* __restrict__ h0, const int* __restrict__ active,
                         float* __restrict__ yf, float* __restrict__ yb)
{
    int e = blockIdx.z; if (!active[e]) return;
    int dir = blockIdx.y, ch = blockIdx.x, d = threadIdx.x;
    const float* A = (dir == 0 ? Af : Ab) + d*DSTATE;
    float Ar[DSTATE];
    #pragma unroll
    for (int n = 0; n < DSTATE; ++n) Ar[n] = A[n];
    float Dv = (dir == 0 ? Df : Db)[d];
    const float* de  = delta + (size_t)e*T_DIM*C_DIM;
    const float* xe  = xi    + (size_t)e*T_DIM*C_DIM;
    const float* bce = bc    + (size_t)e*T_DIM*16;
    float* yo = (dir == 0 ? yf : yb) + (size_t)e*T_DIM*C_DIM;
    const float* hh = h0 + ((((size_t)e*2 + dir)*NCHUNK + ch)*C_DIM + d)*8;
    float h[DSTATE];
    #pragma unroll
    for (int n = 0; n < DSTATE; ++n) h[n] = hh[n];
    for (int i = 0; i < CHUNK; ++i) {
        int p = ch*CHUNK + i;
        int t = dir ? (T_DIM-1-p) : p;
        float dl = de[(size_t)t*C_DIM + d];
        float xv = xe[(size_t)t*C_DIM + d];
        float dx = dl*xv;
        const float* bcr = bce + (size_t)t*16;
        float y = 0.f;
        #pragma unroll
        for (int n = 0; n < DSTATE; ++n) {
            float da = __expf(dl*Ar[n]);
            h[n] = da*h[n] + dx*bcr[n];
            y += h[n]*bcr[8+n];
        }
        yo[(size_t)t*C_DIM + d] = y + xv*Dv;
    }
}

// K6a: R[t] = feats[:,t] . mlp_w   (residual term of the fused logit)
__global__ void k6a_res(const float* __restrict__ feats, const float* __restrict__ mlp_w,
                        const int* __restrict__ active, float* __restrict__ R)
{
    int e = blockIdx.z; if (!active[e]) return;
    int t = blockIdx.x*blockDim.x + threadIdx.x; if (t >= T_DIM) return;
    const float* f = feats + (size_t)e*C_DIM*T_DIM + t;
    float acc = 0.f;
    for (int c = 0; c < C_DIM; ++c) acc += f[(size_t)c*T_DIM]*mlp_w[c];
    R[(size_t)e*T_DIM + t] = acc;
}

// K6b: logit = R + sum_c [0.5*(yf+yb)*silu(z)]*w2 + mlp_b ; mask = logit>0.
//      block = 128 (one t per block), wave-shuffle + LDS reduction.
__global__ void k6b_mask(const float* __restrict__ yf, const float* __restrict__ yb,
                         const float* __restrict__ z, const float* __restrict__ w2,
                         const float* __restrict__ R, const float* __restrict__ mlp_b,
                         const int* __restrict__ active, float* __restrict__ mask)
{
    int e = blockIdx.z; if (!active[e]) return;
    int t = blockIdx.x, c = threadIdx.x;
    size_t o = ((size_t)e*T_DIM + t)*C_DIM + c;
    float y = 0.5f*(yf[o] + yb[o])*siluf_(z[o]);
    float v = y*w2[c];
    #pragma unroll
    for (int off = 16; off > 0; off >>= 1) v += __shfl_down(v, off, 32);
    __shared__ float red[4];
    if ((threadIdx.x & 31) == 0) red[threadIdx.x >> 5] = v;
    __syncthreads();
    if (threadIdx.x == 0) {
        float logit = red[0]+red[1]+red[2]+red[3] + R[(size_t)e*T_DIM + t] + mlp_b[0];
        mask[(size_t)e*T_DIM + t] = (logit > 0.f) ? 1.f : 0.f;
    }
}

// K7: 3x3 clamped max-pool; inactive (ego / singleton) elements are all ones.
__global__ void k7_pool(const float* __restrict__ mask, const int* __restrict__ active,
                        float* __restrict__ out)
{
    int e = blockIdx.z;
    int idx = blockIdx.x*blockDim.x + threadIdx.x; if (idx >= T_DIM) return;
    float v;
    if (!active[e]) v = 1.f;
    else {
        int h = idx/WW, w = idx - h*WW;
        v = -INFINITY;
        const float* m = mask + (size_t)e*T_DIM;
        for (int dh = -1; dh <= 1; ++dh) {
            int hh = h + dh; if (hh < 0 || hh >= HH) continue;
            for (int dw = -1; dw <= 1; ++dw) {
                int ww = w + dw; if (ww < 0 || ww >= WW) continue;
                v = fmaxf(v, m[hh*WW + ww]);
            }
        }
    }
    out[(size_t)e*T_DIM + idx] = v;
}

// ---------------------------------------------------------------------------
extern "C" void kernel_launch(void* const* d_in, const int* in_sizes, int n_in,
                              void* d_out, int out_size, void* d_ws, size_t ws_size,
                              hipStream_t stream)
{
    const float* feats      = (const float*)d_in[0];
    const float* in_proj    = (const float*)d_in[1];
    const float* conv_w     = (const float*)d_in[2];
    const float* conv_b     = (const float*)d_in[3];
    const float* x_proj_w   = (const float*)d_in[4];
    const float* dt_proj_w  = (const float*)d_in[5];
    const float* dt_proj_b  = (const float*)d_in[6];
    const float* A_log_f    = (const float*)d_in[7];
    const float* A_log_b    = (const float*)d_in[8];
    const float* D_f        = (const float*)d_in[9];
    const float* D_b        = (const float*)d_in[10];
    const float* out_proj_w = (const float*)d_in[11];
    // rms_w = d_in[12] used in k0
    const float* mlp_w      = (const float*)d_in[13];
    const float* mlp_b      = (const float*)d_in[14];
    const int*   rec        = (const int*)d_in[15];
    const float* rms_w      = (const float*)d_in[12];

    const int N       = in_sizes[0] / (C_DIM*T_DIM);
    const int ngroups = in_sizes[15];
    float* out = (float*)d_out;

    // workspace carve-out (float units, 64-float aligned)
    float* ws = (float*)d_ws;
    size_t off = 0;
    auto carve = [&](size_t n) { float* p = ws + off; off += (n + 63) & ~(size_t)63; return p; };

    int*   active = (int*)carve(64);
    float* Af   = carve(C_DIM*DSTATE);
    float* Ab   = carve(C_DIM*DSTATE);
    float* Wip  = carve(C_DIM*256);
    float* Wc   = carve(C_DIM*144);
    float* w2   = carve(C_DIM);
    float* sS   = carve((size_t)N*T_DIM);
    float* R    = carve((size_t)N*T_DIM);
    float* maskB= carve((size_t)N*T_DIM);
    float* yfB  = carve((size_t)N*T_DIM*C_DIM);
    float* ybB  = carve((size_t)N*T_DIM*C_DIM);  // holds xi0 before conv, yb after
    float* zB   = carve((size_t)N*T_DIM*C_DIM);
    float* dlB  = carve((size_t)N*T_DIM*C_DIM);
    float* xiB  = carve((size_t)N*T_DIM*C_DIM);
    float* bcB  = carve((size_t)N*T_DIM*16);
    float* part = carve((size_t)N*2*NCHUNK*C_DIM*16);
    float* h0   = carve((size_t)N*2*NCHUNK*C_DIM*8);
    (void)ws_size; (void)n_in; (void)out_size;

    k0_precomp<<<dim3(1), dim3(256), 0, stream>>>(rec, ngroups, N, in_proj, rms_w,
        x_proj_w, dt_proj_w, out_proj_w, mlp_w, A_log_f, A_log_b,
        active, Wip, Wc, w2, Af, Ab);

    k1_rms<<<dim3(T_DIM/256, 1, N), dim3(256), 0, stream>>>(feats, active, sS);

    k2_gemm1<<<dim3(4, 36, N), dim3(128), 0, stream>>>(feats, Wip, sS, active, ybB, zB);

    k3_conv<<<dim3(T_DIM/CHUNK, 1, N), dim3(C_DIM), 0, stream>>>(ybB, conv_w, conv_b, active, xiB);

    k4_gemm2<<<dim3(9, 36, N), dim3(32), 0, stream>>>(xiB, Wc, dt_proj_b, active, dlB, bcB);

    k5a_part<<<dim3(NCHUNK, 2, N), dim3(C_DIM), 0, stream>>>(dlB, xiB, bcB, Af, Ab, active, part);
    k5b_carry<<<dim3(2, N), dim3(C_DIM), 0, stream>>>(active, part, h0);
    k5c_scan<<<dim3(NCHUNK, 2, N), dim3(C_DIM), 0, stream>>>(dlB, xiB, bcB, Af, Ab,
        D_f, D_b, h0, active, yfB, ybB);

    k6a_res<<<dim3(T_DIM/256, 1, N), dim3(256), 0, stream>>>(feats, mlp_w, active, R);
    k6b_mask<<<dim3(T_DIM, 1, N), dim3(C_DIM), 0, stream>>>(yfB, ybB, zB, w2, R, mlp_b, active, maskB);

    k7_pool<<<dim3((T_DIM+255)/256, 1, N), dim3(256), 0, stream>>>(maskB, active, out);
}